// TensorProductScoreModel_57097295233128
// MI455X (gfx1250) — compile-verified
//
#include <hip/hip_runtime.h>

// ---------------- problem constants (from reference) ----------------
#define NSC 32            // ns
#define NVC 8             // nv
#define N_NODES 10000
#define N_EDGES 160000
#define EDGE_FEAT 96      // 3*ns
#define SH_DIM 9
#define W_NUMEL 1280      // 32*32 + 32*8
#define OUT_DIM 56        // 32 + 3*8
#define TILE_E 32         // edges per block = two WMMA M-tiles (B-fragment reuse)
#define EA_STRIDE 100     // padded LDS row stride (even -> float2-aligned)

typedef __attribute__((ext_vector_type(2))) float v2f;
typedef __attribute__((ext_vector_type(8))) float v8f;

// V_WMMA_F32_16X16X4_F32 : D = A(16x4) * B(4x16) + C(16x16), exact fp32
__device__ __forceinline__ v8f wmma4(v2f a, v2f b, v8f c) {
  return __builtin_amdgcn_wmma_f32_16x16x4_f32(
      /*neg_a=*/false, a, /*neg_b=*/false, b,
      /*c_mod=*/(short)0, c, /*reuse_a=*/false, /*reuse_b=*/false);
}

__global__ void zero_ws_kernel(float* __restrict__ p, int n) {
  int i = blockIdx.x * 256 + threadIdx.x;
  if (i < n) p[i] = 0.f;
}

__global__ __launch_bounds__(256)
void tpcl_edge_kernel(const float* __restrict__ node_attr,
                      const int*   __restrict__ edge_index,
                      const float* __restrict__ edge_attr,
                      const float* __restrict__ edge_sh,
                      const float* __restrict__ fc_w1,
                      const float* __restrict__ fc_b1,
                      const float* __restrict__ fc_w2,
                      const float* __restrict__ fc_b2,
                      float* __restrict__ sums,
                      float* __restrict__ cnts)
{
  __shared__ float Ea[TILE_E][EA_STRIDE];   // edge_attr tile (2 M-tiles)
  __shared__ float H [TILE_E][EA_STRIDE];   // relu(FC1) tile
  __shared__ float Xs[TILE_E][NSC];         // gathered node_attr[dst]
  __shared__ float O0[TILE_E][NSC];         // sum_u x_u * w0[u,:]
  __shared__ float O1[TILE_E][NVC];         // sum_u x_u * w1[u,:]
  __shared__ float SHs[TILE_E][12];
  __shared__ int   SRC[TILE_E];
  __shared__ int   DST[TILE_E];

  const int tid  = threadIdx.x;
  const int lane = tid & 31;
  const int wid  = tid >> 5;               // 8 wave32s
  const int e0   = blockIdx.x * TILE_E;

  // ---- Phase A: stage inputs, zero accumulators ----
  if (tid < TILE_E) {
    SRC[tid] = edge_index[e0 + tid];              // edge_index[0, e]
    DST[tid] = edge_index[N_EDGES + e0 + tid];    // edge_index[1, e]
  }
  for (int i = tid; i < TILE_E * EDGE_FEAT; i += 256) {
    int r = i / EDGE_FEAT, c = i - r * EDGE_FEAT;
    Ea[r][c] = edge_attr[(size_t)(e0 + r) * EDGE_FEAT + c];
  }
  for (int i = tid; i < TILE_E * SH_DIM; i += 256) {
    int r = i / SH_DIM, c = i - r * SH_DIM;
    SHs[r][c] = edge_sh[(size_t)(e0 + r) * SH_DIM + c];
  }
  for (int i = tid; i < TILE_E * NSC; i += 256) O0[i >> 5][i & 31] = 0.f;
  for (int i = tid; i < TILE_E * NVC; i += 256) O1[i >> 3][i & 7] = 0.f;
  __syncthreads();

  // ---- Phase B: gather x = node_attr[dst]; FC1 via WMMA on waves 0..5 ----
  for (int i = tid; i < TILE_E * NSC; i += 256) {
    int r = i >> 5, c = i & 31;
    Xs[r][c] = node_attr[(size_t)DST[r] * NSC + c];
  }

  // WMMA fp32 fragment addressing (ISA 7.12.2):
  //  A 16x4: lane holds row m=lane&15; vgpr pair = (K, K+1), lanes 16..31 at K+2
  //  B 4x16: lane holds col n=lane&15; vgpr pair = rows (K, K+1), +2 for hi lanes
  //  C 16x16: lane holds col n=lane&15; vgpr j holds row m = j + 8*(lane>>4)
  const int m    = lane & 15;
  const int kb   = (lane >> 4) << 1;
  const int cl   = lane & 15;
  const int mrow = (lane >> 4) << 3;

  if (wid < 6) {                       // FC1: N = 96 -> 6 tiles of 16; 2 M-tiles
    const int ncol = (wid << 4) + cl;
    v8f acc_a = {};                    // edges 0..15
    v8f acc_b = {};                    // edges 16..31
    #pragma unroll 4
    for (int k = 0; k < EDGE_FEAT; k += 4) {
      v2f a0, a1, b;
      a0.x = Ea[m][k + kb];
      a0.y = Ea[m][k + kb + 1];
      a1.x = Ea[16 + m][k + kb];
      a1.y = Ea[16 + m][k + kb + 1];
      b.x = fc_w1[(size_t)(k + kb)     * EDGE_FEAT + ncol];
      b.y = fc_w1[(size_t)(k + kb + 1) * EDGE_FEAT + ncol];
      acc_a = wmma4(a0, b, acc_a);     // B fragment reused across both M-tiles
      acc_b = wmma4(a1, b, acc_b);
    }
    const float bias = fc_b1[ncol];
    #pragma unroll
    for (int j = 0; j < 8; ++j) {
      float va = acc_a[j] + bias;
      float vb = acc_b[j] + bias;
      H[mrow + j][ncol]      = va > 0.f ? va : 0.f;   // ReLU
      H[16 + mrow + j][ncol] = vb > 0.f ? vb : 0.f;
    }
  }
  __syncthreads();

  // ---- Stage 2: FC2 (80 N-tiles of 16) fused with contraction against x ----
  // Tiles t<64: w0 block -> single u = t>>1, out0 columns (t&1)*16 + cl.
  // All tiles of a wave share parity (t = wid + 8s), so out0 accumulates in regs.
  v8f acc0a = {};
  v8f acc0b = {};
  for (int s = 0; s < 10; ++s) {
    const int t    = wid + (s << 3);
    const int ncol = (t << 4) + cl;
    v8f c0 = {};
    v8f c1 = {};
    #pragma unroll 4
    for (int k = 0; k < EDGE_FEAT; k += 4) {
      v2f a0, a1, b;
      a0.x = H[m][k + kb];
      a0.y = H[m][k + kb + 1];
      a1.x = H[16 + m][k + kb];
      a1.y = H[16 + m][k + kb + 1];
      b.x = fc_w2[(size_t)(k + kb)     * W_NUMEL + ncol];
      b.y = fc_w2[(size_t)(k + kb + 1) * W_NUMEL + ncol];
      c0 = wmma4(a0, b, c0);           // shared B fragment: 2 WMMA per 2 loads
      c1 = wmma4(a1, b, c1);
    }
    const float bias = fc_b2[ncol];
    if (t < 64) {
      const int u = t >> 1;                          // w0: col = u*32 + c
      #pragma unroll
      for (int j = 0; j < 8; ++j) {
        acc0a[j] += (c0[j] + bias) * Xs[mrow + j][u];
        acc0b[j] += (c1[j] + bias) * Xs[16 + mrow + j][u];
      }
    } else {
      const int u = ((t - 64) << 1) + (cl >> 3);     // w1: col-1024 = u*8 + v
      const int v = cl & 7;
      #pragma unroll
      for (int j = 0; j < 8; ++j) {
        atomicAdd(&O1[mrow + j][v],      (c0[j] + bias) * Xs[mrow + j][u]);
        atomicAdd(&O1[16 + mrow + j][v], (c1[j] + bias) * Xs[16 + mrow + j][u]);
      }
    }
  }
  {
    const int chalf = (wid & 1) << 4;   // 4 waves share each 16-col half -> LDS atomics
    #pragma unroll
    for (int j = 0; j < 8; ++j) {
      atomicAdd(&O0[mrow + j][chalf + cl],      acc0a[j]);
      atomicAdd(&O0[16 + mrow + j][chalf + cl], acc0b[j]);
    }
  }
  __syncthreads();

  // ---- Epilogue: fan-in norm, SH scaling, scatter-add to src nodes ----
  const float inv = 0.17677669529663688f;  // 1/sqrt(32)
  for (int i = tid; i < TILE_E * OUT_DIM; i += 256) {
    int e = i / OUT_DIM, c = i - e * OUT_DIM;
    float val;
    if (c < NSC) {
      val = inv * O0[e][c] * SHs[e][0];
    } else {
      int rel = c - NSC;
      int v = rel / 3, d = rel - v * 3;    // flatten [NV,3] -> v*3+d
      val = inv * O1[e][v] * SHs[e][1 + d];
    }
    atomicAdd(&sums[(size_t)SRC[e] * OUT_DIM + c], val);
  }
  if (tid < TILE_E) atomicAdd(&cnts[SRC[tid]], 1.0f);
}

__global__ void finalize_kernel(const float* __restrict__ sums,
                                const float* __restrict__ cnts,
                                float* __restrict__ out)
{
  int i = blockIdx.x * 256 + threadIdx.x;
  if (i < N_NODES * OUT_DIM) {
    float c = cnts[i / OUT_DIM];
    out[i] = sums[i] / (c > 1.f ? c : 1.f);   // scatter-mean
  }
}

extern "C" void kernel_launch(void* const* d_in, const int* in_sizes, int n_in,
                              void* d_out, int out_size, void* d_ws, size_t ws_size,
                              hipStream_t stream) {
  const float* node_attr  = (const float*)d_in[0];
  const int*   edge_index = (const int*)  d_in[1];
  const float* edge_attr  = (const float*)d_in[2];
  const float* edge_sh    = (const float*)d_in[3];
  const float* fc_w1      = (const float*)d_in[4];
  const float* fc_b1      = (const float*)d_in[5];
  const float* fc_w2      = (const float*)d_in[6];
  const float* fc_b2      = (const float*)d_in[7];

  float* sums = (float*)d_ws;                         // [N_NODES * OUT_DIM]
  float* cnts = sums + (size_t)N_NODES * OUT_DIM;     // [N_NODES]

  const int ztotal = N_NODES * OUT_DIM + N_NODES;
  zero_ws_kernel<<<(ztotal + 255) / 256, 256, 0, stream>>>(sums, ztotal);

  tpcl_edge_kernel<<<N_EDGES / TILE_E, 256, 0, stream>>>(
      node_attr, edge_index, edge_attr, edge_sh,
      fc_w1, fc_b1, fc_w2, fc_b2, sums, cnts);

  finalize_kernel<<<(N_NODES * OUT_DIM + 255) / 256, 256, 0, stream>>>(
      sums, cnts, (float*)d_out);
}